// KalmanFilter_55198919688454
// MI455X (gfx1250) — compile-verified
//
#include <hip/hip_runtime.h>
#include <math.h>

// ---------------------------------------------------------------------------
// Kalman filter on MI455X (gfx1250), wave32.
//
// Packing: each wave processes 2 batch elements as block-diagonal 16x16 tiles
//   (batch b0 -> rows/cols 0-7, batch b1 -> rows/cols 8-15).
// All dense products use V_WMMA_F32_16X16X4_F32 (D = A(16x4) x B(4x16) + C),
// chained over K-chunks. C/D layout (ISA 7.12.2): VGPR r, lane l:
//   element (M = r + 8*(l>=16), N = l&15).
// A layout (16x4 f32): VGPR v, lane l: element (M = l&15, K = 4c + v + 2*(l>>4)).
// B layout (4x16 f32): assumed mirror of A: VGPR v, lane l:
//   element (K = 4c + v + 2*(l>>4), N = l&15).
// Layout conversions go through a per-wave LDS tile stored column-major:
//   element(row m, col n) at tile[n*16 + m].
// ---------------------------------------------------------------------------

typedef __attribute__((ext_vector_type(2))) float v2f;
typedef __attribute__((ext_vector_type(8))) float v8f;

#define LDS_FENCE() asm volatile("s_wait_dscnt 0" ::: "memory")

#define WAVES_PER_BLOCK 8
#define SLICE 448  // per-wave floats: tile 256 + xtile 128 + mvec 16 + rvec 8 (pad)

__device__ __forceinline__ v8f wmma4(v2f a, v2f b, v8f c) {
  // D = A x B + C, fp32 16x16x4
  return __builtin_amdgcn_wmma_f32_16x16x4_f32(false, a, false, b, (short)0, c,
                                               false, false);
}

// Load A-operand k-chunk c of a 16x16 matrix held column-major in `t`:
// A[m,k] = t[k*16 + m]
__device__ __forceinline__ v2f loadA(const float* t, int c, int ln, int half) {
  const int k0 = 4 * c + 2 * half;
  v2f a;
  a[0] = t[(k0 + 0) * 16 + ln];
  a[1] = t[(k0 + 1) * 16 + ln];
  return a;
}

// Load B-operand k-chunk c: B[k,n] = t[n*16 + k]
__device__ __forceinline__ v2f loadB(const float* t, int c, int ln, int half) {
  const int k0 = 4 * c + 2 * half;
  v2f b;
  b[0] = t[ln * 16 + k0 + 0];
  b[1] = t[ln * 16 + k0 + 1];
  return b;
}

// Store a C/D-layout tile (v8f per lane) to LDS column-major.
__device__ __forceinline__ void storeTileD(float* t, int ln, int half, v8f d) {
  float4* p = (float4*)(t + ln * 16 + 8 * half);  // 32B aligned
  p[0] = make_float4(d[0], d[1], d[2], d[3]);
  p[1] = make_float4(d[4], d[5], d[6], d[7]);
}

__global__ __launch_bounds__(WAVES_PER_BLOCK * 32) void kf_kernel(
    const float* __restrict__ emissions, const float* __restrict__ m0g,
    const float* __restrict__ P0g, const float* __restrict__ Ag,
    const float* __restrict__ Qg, const float* __restrict__ Hg,
    const float* __restrict__ Rg, float* __restrict__ out_ll,
    float* __restrict__ out_means, float* __restrict__ out_covs, int B, int T) {
  __shared__ float smem[WAVES_PER_BLOCK * SLICE];

  const int tid = threadIdx.x;
  const int wave = tid >> 5;
  const int lane = tid & 31;
  const int half = lane >> 4;   // which batch of the pair for replicated work
  const int ln = lane & 15;     // N / M index within the tile
  const int pair = blockIdx.x * WAVES_PER_BLOCK + wave;
  const int b0 = pair * 2;
  if (b0 >= B) return;
  const int bg = b0 + half;     // this lane's global batch id

  float* tile = smem + wave * SLICE;  // 16x16 tile, column-major
  float* xt = tile + 256;             // 8x16 X tile, column-major stride 8
  float* mv = tile + 384;             // mean vectors [2][8]
  float* rv = tile + 400;             // residuals   [2][4]

  // ---- constant operands, built once (block-diagonal packing) ------------
  v2f AblkA[4], AblkTB[4], HblkA[4], HblkTB[4];
#pragma unroll
  for (int c = 0; c < 4; ++c) {
#pragma unroll
    for (int v = 0; v < 2; ++v) {
      const int K = 4 * c + 2 * half + v;
      const int M = ln, N = ln;
      const bool sbA = (M >> 3) == (K >> 3);
      const bool sbB = (N >> 3) == (K >> 3);
      AblkA[c][v] = sbA ? Ag[(M & 7) * 8 + (K & 7)] : 0.f;            // Ablk[M,K]
      AblkTB[c][v] = sbB ? Ag[(N & 7) * 8 + (K & 7)] : 0.f;           // Ablk^T[K,N]
      HblkA[c][v] = (sbA && (M & 7) < 4) ? Hg[(M & 3) * 8 + (K & 7)] : 0.f;
      HblkTB[c][v] = (sbB && (N & 7) < 4) ? Hg[(N & 3) * 8 + (K & 7)] : 0.f;
    }
  }
  v8f Qd, Rd;
  {
    const int nblk = ln >> 3, j = ln & 7;
#pragma unroll
    for (int r = 0; r < 8; ++r) {
      Qd[r] = (half == nblk) ? Qg[r * 8 + j] : 0.f;
      Rd[r] = (half == nblk && r < 4 && j < 4) ? Rg[(r & 3) * 4 + (j & 3)] : 0.f;
    }
  }
  float Arow[8], Hrow[8], mrep[8];
#pragma unroll
  for (int j = 0; j < 8; ++j) {
    Arow[j] = Ag[(lane & 7) * 8 + j];  // row (lane&7) of A
    Hrow[j] = Hg[(lane & 3) * 8 + j];  // row (lane&3) of H
    mrep[j] = m0g[(size_t)bg * 8 + j]; // replicated filtered mean
  }
  // initial covariance tile (block-diagonal)
  v8f P;
  {
    const int nblk = ln >> 3, j = ln & 7;
#pragma unroll
    for (int r = 0; r < 8; ++r)
      P[r] = (half == nblk) ? P0g[(size_t)bg * 64 + r * 8 + j] : 0.f;
  }

  const int d = lane & 7;                            // column role
  const bool activeCol = (lane < 8) || (lane >= 24); // column lanes
  const float LN2PI = 1.8377997f + 7.735e-8f;        // log(2*pi)
  float ll_acc = 0.f;
  const v8f zero8 = {0.f, 0.f, 0.f, 0.f, 0.f, 0.f, 0.f, 0.f};

  for (int t = 0; t < T; ++t) {
    // ---- T1 = Ablk * P  (K=16, 4 WMMAs) -------------------------------
    storeTileD(tile, ln, half, P);
    LDS_FENCE();
    v8f acc = zero8;
#pragma unroll
    for (int c = 0; c < 4; ++c) acc = wmma4(AblkA[c], loadB(tile, c, ln, half), acc);

    // ---- Pp = T1 * Ablk^T + Qblk --------------------------------------
    storeTileD(tile, ln, half, acc);
    LDS_FENCE();
    v8f Pp = Qd;
#pragma unroll
    for (int c = 0; c < 4; ++c) Pp = wmma4(loadA(tile, c, ln, half), AblkTB[c], Pp);

    // ---- HPp = Hblk * Pp ----------------------------------------------
    storeTileD(tile, ln, half, Pp);
    LDS_FENCE();
    acc = zero8;
#pragma unroll
    for (int c = 0; c < 4; ++c) acc = wmma4(HblkA[c], loadB(tile, c, ln, half), acc);

    // ---- stage HPp; pull A-operands + per-column hp before overwrite ---
    storeTileD(tile, ln, half, acc);
    LDS_FENCE();
    v2f HSa[4], HPf[2];
#pragma unroll
    for (int c = 0; c < 4; ++c) HSa[c] = loadA(tile, c, ln, half);
#pragma unroll
    for (int c = 0; c < 2; ++c)
#pragma unroll
      for (int v = 0; v < 2; ++v) {
        const int k = 4 * c + 2 * half + v;  // blockified HPp^T: A[m,k]=HPp[(k&3)+8*(k>>2), m]
        HPf[c][v] = tile[ln * 16 + (k & 3) + ((k >> 2) << 3)];
      }
    float hp[4];
#pragma unroll
    for (int e = 0; e < 4; ++e)
      hp[e] = tile[(d + 8 * half) * 16 + (e + 8 * half)];  // HPp[:, this column]

    // ---- S = HPp * Hblk^T + Rblk --------------------------------------
    v8f S = Rd;
#pragma unroll
    for (int c = 0; c < 4; ++c) S = wmma4(HSa[c], HblkTB[c], S);

    storeTileD(tile, ln, half, S);
    LDS_FENCE();
    const int off = half * (8 * 16 + 8);  // S(i,j) at tile[off + j*16 + i]
    const float s00 = tile[off + 0 * 16 + 0], s10 = tile[off + 0 * 16 + 1];
    const float s20 = tile[off + 0 * 16 + 2], s30 = tile[off + 0 * 16 + 3];
    const float s11 = tile[off + 1 * 16 + 1], s21 = tile[off + 1 * 16 + 2];
    const float s31 = tile[off + 1 * 16 + 3], s22 = tile[off + 2 * 16 + 2];
    const float s32 = tile[off + 2 * 16 + 3], s33 = tile[off + 3 * 16 + 3];

    // ---- mean predict: mp = A m (row-distributed, then broadcast) ------
    float mp_d = 0.f;
#pragma unroll
    for (int j = 0; j < 8; ++j) mp_d = fmaf(Arow[j], mrep[j], mp_d);
    if (activeCol) mv[half * 8 + d] = mp_d;
    LDS_FENCE();
    float mpr[8];
#pragma unroll
    for (int j = 0; j < 8; ++j) mpr[j] = mv[half * 8 + j];

    // ---- innovation r = y - H mp --------------------------------------
    float mu = 0.f;
#pragma unroll
    for (int j = 0; j < 8; ++j) mu = fmaf(Hrow[j], mpr[j], mu);
    const float y = emissions[((size_t)bg * T + t) * 4 + (lane & 3)];
    const float rres = y - mu;
    if (activeCol && d < 4) rv[half * 4 + d] = rres;
    LDS_FENCE();
    const float rr0 = rv[half * 4 + 0], rr1 = rv[half * 4 + 1];
    const float rr2 = rv[half * 4 + 2], rr3 = rv[half * 4 + 3];

    // ---- Cholesky of S (replicated per lane) --------------------------
    const float L00 = sqrtf(s00), i00 = 1.f / L00;
    const float L10 = s10 * i00, L20 = s20 * i00, L30 = s30 * i00;
    const float L11 = sqrtf(s11 - L10 * L10), i11 = 1.f / L11;
    const float L21 = (s21 - L20 * L10) * i11;
    const float L31 = (s31 - L30 * L10) * i11;
    const float L22 = sqrtf(s22 - L20 * L20 - L21 * L21), i22 = 1.f / L22;
    const float L32 = (s32 - L30 * L20 - L31 * L21) * i22;
    const float L33 = sqrtf(s33 - L30 * L30 - L31 * L31 - L32 * L32), i33 = 1.f / L33;

    // ---- log-likelihood -----------------------------------------------
    const float z0 = rr0 * i00;
    const float z1 = (rr1 - L10 * z0) * i11;
    const float z2 = (rr2 - L20 * z0 - L21 * z1) * i22;
    const float z3 = (rr3 - L30 * z0 - L31 * z1 - L32 * z2) * i33;
    ll_acc += -0.5f * (z0 * z0 + z1 * z1 + z2 * z2 + z3 * z3) -
              (logf(L00) + logf(L11) + logf(L22) + logf(L33)) - 2.f * LN2PI;

    // ---- X = -S^{-1} HPp (per column; fwd+back substitution) ----------
    const float w0 = hp[0] * i00;
    const float w1 = (hp[1] - L10 * w0) * i11;
    const float w2 = (hp[2] - L20 * w0 - L21 * w1) * i22;
    const float w3 = (hp[3] - L30 * w0 - L31 * w1 - L32 * w2) * i33;
    const float x3 = w3 * i33;
    const float x2 = (w2 - L32 * x3) * i22;
    const float x1 = (w1 - L21 * x2 - L31 * x3) * i11;
    const float x0 = (w0 - L10 * x1 - L20 * x2 - L30 * x3) * i00;
    const float xp0 = -x0, xp1 = -x1, xp2 = -x2, xp3 = -x3;

    if (activeCol) {  // build 8x16 block-diag X tile (zeros in cross block)
      const int n = d + 8 * half;
      *(float4*)(xt + n * 8 + 4 * half) = make_float4(xp0, xp1, xp2, xp3);
      *(float4*)(xt + n * 8 + 4 * (1 - half)) = make_float4(0.f, 0.f, 0.f, 0.f);
    }
    LDS_FENCE();
    v2f XB0, XB1;
#pragma unroll
    for (int v = 0; v < 2; ++v) {
      XB0[v] = xt[ln * 8 + (2 * half + v)];
      XB1[v] = xt[ln * 8 + (4 + 2 * half + v)];
    }

    // ---- Pf = Pp - HPp^T S^{-1} HPp  (K=8, 2 WMMAs) -------------------
    v8f Pf = wmma4(HPf[0], XB0, Pp);
    Pf = wmma4(HPf[1], XB1, Pf);
    P = Pf;

    // ---- mf = mp + K r  (K r = -sum_e xp_e * r_e) ---------------------
    const float mf_d =
        mp_d - (xp0 * rr0 + xp1 * rr1 + xp2 * rr2 + xp3 * rr3);

    if (activeCol) {
      mv[half * 8 + d] = mf_d;
      out_means[((size_t)bg * T + t) * 8 + d] = mf_d;
      // Pf symmetric: lane's column d == row d -> contiguous b128 stores
      float* cp = out_covs + (((size_t)bg * T + t) * 64 + d * 8);
      *(float4*)(cp + 0) = make_float4(Pf[0], Pf[1], Pf[2], Pf[3]);
      *(float4*)(cp + 4) = make_float4(Pf[4], Pf[5], Pf[6], Pf[7]);
    }
    LDS_FENCE();
#pragma unroll
    for (int j = 0; j < 8; ++j) mrep[j] = mv[half * 8 + j];
  }

  if ((lane & 15) == 0) out_ll[bg] = ll_acc;  // lanes 0 & 16 -> b0, b1
}

extern "C" void kernel_launch(void* const* d_in, const int* in_sizes, int n_in,
                              void* d_out, int out_size, void* d_ws,
                              size_t ws_size, hipStream_t stream) {
  (void)n_in; (void)out_size; (void)d_ws; (void)ws_size;
  const float* emissions = (const float*)d_in[0];
  const float* m0 = (const float*)d_in[1];
  const float* P0 = (const float*)d_in[2];
  const float* A = (const float*)d_in[3];
  const float* Q = (const float*)d_in[4];
  const float* H = (const float*)d_in[5];
  const float* R = (const float*)d_in[6];

  const int B = in_sizes[1] / 8;            // m0 is [B, D=8]
  const int T = in_sizes[0] / (B * 4);      // emissions [B, T, E=4]

  float* out_ll = (float*)d_out;            // [B]
  float* out_means = out_ll + B;            // [B, T, 8]
  float* out_covs = out_means + (size_t)B * T * 8;  // [B, T, 8, 8]

  const int pairs = (B + 1) / 2;
  const int blocks = (pairs + WAVES_PER_BLOCK - 1) / WAVES_PER_BLOCK;
  kf_kernel<<<blocks, WAVES_PER_BLOCK * 32, 0, stream>>>(
      emissions, m0, P0, A, Q, H, R, out_ll, out_means, out_covs, B, T);
}